// ResiduePooling_16045997818006
// MI455X (gfx1250) — compile-verified
//
#include <hip/hip_runtime.h>
#include <cstdint>

// Scatter-mean over sorted segments:
//   out[r, :] = sum_{i : ridx[i]==r} feat[i, :] / max(count_r, 1)
//
// MI455X plan: memory-bound (AI ~0.25 flop/byte, ~50us floor @ 23.3 TB/s).
// One wave32 per residue; wave-uniform (scalar) binary search on the sorted
// index; stage whole 512B feature rows into LDS with
// global_load_async_to_lds_b128 th:TH_LOAD_NT (1 instruction == 1 row,
// fully coalesced, no L2 pollution), s_wait_asynccnt, accumulate via
// ds_load_b128, one non-temporal coalesced b128 store.

typedef float v4f __attribute__((ext_vector_type(4)));  // native vector: NT-builtin friendly

#define WAVES_PER_BLOCK 8
#define BLOCK_THREADS   (WAVES_PER_BLOCK * 32)
#define DEPTH           8              // staged rows per wave per batch
#define D               128            // feature dim
#define ROW_BYTES       (D * 4)        // 512 B per atom row
#define SMEM_BYTES      (WAVES_PER_BLOCK * DEPTH * ROW_BYTES)  // 32 KB

// Wave-uniform lower bound: 'val' is uniform (readfirstlane'd by caller), so
// the compiler can keep lo/hi/mid in SGPRs and use scalar loads.
__device__ __forceinline__ int lower_bound_i32(const int* __restrict__ a,
                                               int lo, int hi, int val) {
    while (lo < hi) {
        int mid = (lo + hi) >> 1;
        if (a[mid] < val) lo = mid + 1; else hi = mid;
    }
    return lo;
}

__global__ __launch_bounds__(BLOCK_THREADS)
void residue_pool_mean_kernel(const float* __restrict__ feat,
                              const int*   __restrict__ ridx,
                              float*       __restrict__ out,
                              int n_atoms, int n_res) {
    extern __shared__ v4f smem4[];      // [WAVES_PER_BLOCK][DEPTH][32] v4f

    const int lane = threadIdx.x & 31;
    const int wave = threadIdx.x >> 5;
    const int r    = blockIdx.x * WAVES_PER_BLOCK + wave;
    if (r >= n_res) return;

    // r is identical across all 32 lanes of this wave; make that explicit so
    // the segment search lowers to SALU + scalar loads.
    const int ru = __builtin_amdgcn_readfirstlane(r);

    const int seg_beg = lower_bound_i32(ridx, 0, n_atoms, ru);
    const int seg_end = lower_bound_i32(ridx, seg_beg, n_atoms, ru + 1);

    const unsigned wave_lds_base = (unsigned)wave * (DEPTH * ROW_BYTES);
    const int      wave_slot0    = wave * DEPTH * 32;   // v4f index into smem4
    const uint64_t feat_base     = (uint64_t)(uintptr_t)feat;
    const unsigned lane_off      = (unsigned)lane * 16u;

    v4f acc = {0.f, 0.f, 0.f, 0.f};

    for (int b = seg_beg; b < seg_end; b += DEPTH) {
        const int cnt = min(DEPTH, seg_end - b);

        // Async DMA: one instruction moves one full 512B atom row
        // (32 lanes x 16B) global -> this wave's LDS slot. Non-temporal:
        // the feature stream is read exactly once.
        for (int j = 0; j < cnt; ++j) {
            unsigned lds_addr = wave_lds_base + (unsigned)j * ROW_BYTES + lane_off;
            uint64_t gaddr = feat_base + (uint64_t)(b + j) * ROW_BYTES + lane_off;
            asm volatile("global_load_async_to_lds_b128 %0, %1, off th:TH_LOAD_NT"
                         :: "v"(lds_addr), "v"(gaddr)
                         : "memory");
        }
        // Wait for this wave's outstanding async LDS transfers.
        asm volatile("s_wait_asynccnt 0x0" ::: "memory");

        // Accumulate staged rows from LDS (ds_load_b128 per row per lane).
        for (int j = 0; j < cnt; ++j) {
            acc += smem4[wave_slot0 + j * 32 + lane];
        }
    }

    // Mean with count clamped to 1 (empty residues -> zeros).
    const float c = (float)max(seg_end - seg_beg, 1);
    acc /= c;

    // Coalesced non-temporal 512B row store (output is write-once).
    __builtin_nontemporal_store(acc, (v4f*)out + (size_t)r * (D / 4) + lane);
}

extern "C" void kernel_launch(void* const* d_in, const int* in_sizes, int n_in,
                              void* d_out, int out_size, void* d_ws, size_t ws_size,
                              hipStream_t stream) {
    const float* feat = (const float*)d_in[0];   // [n_atoms, 128] f32
    const int*   ridx = (const int*)d_in[1];     // [n_atoms] sorted residue ids
    float*       out  = (float*)d_out;           // [n_res, 128] f32

    const int n_atoms = in_sizes[1];
    const int n_res   = out_size / D;
    const int blocks  = (n_res + WAVES_PER_BLOCK - 1) / WAVES_PER_BLOCK;

    hipLaunchKernelGGL(residue_pool_mean_kernel,
                       dim3(blocks), dim3(BLOCK_THREADS), SMEM_BYTES, stream,
                       feat, ridx, out, n_atoms, n_res);
}